// TimeSeriesHead_66838281061130
// MI455X (gfx1250) — compile-verified
//
#include <hip/hip_runtime.h>

typedef __attribute__((ext_vector_type(2))) float v2f;
typedef __attribute__((ext_vector_type(8))) float v8f;

#define EMBED   768
#define HEADS   300
#define OUTF    3
#define BATCH   32
#define NHALF   384     // output columns of layer-1 per block
#define KC      64      // K staging chunk for f
#define KCP     66      // padded LDS stride for f chunk (bank-conflict free)
#define HSTR    386     // padded LDS stride for h buffer

// ---------------------------------------------------------------------------
// Kernel A: for head o and N-half nh, compute h[32, 384] = relu(f @ W1[o] + b1),
// then partial y[32,3] over those 384 columns -> ws[nh][b][d][o].
// ---------------------------------------------------------------------------
__launch_bounds__(256)
__global__ void head_mlp_kernel(const float* __restrict__ f,
                                const float* __restrict__ W1,
                                const float* __restrict__ b1,
                                const float* __restrict__ W2,
                                float* __restrict__ ws) {
    __shared__ float fchunk[BATCH * KCP];   // 32*66*4  = 8448 B
    __shared__ float hbuf[BATCH * HSTR];    // 32*386*4 = 49408 B

    const int o  = blockIdx.y;
    const int nh = blockIdx.x;            // 0 or 1
    const int t  = threadIdx.x;
    const int wave = t >> 5;
    const int lane = t & 31;
    const int lo = lane & 15;             // lane within half-wave
    const int hi = lane >> 4;             // half-wave index

    const int mtile = wave & 1;           // M tile: rows [mtile*16, +16)
    const int ngrp  = wave >> 1;          // 4 groups x 96 columns
    const int colloc = ngrp * 96;         // local column base (0..383)
    const int colglb = nh * NHALF + colloc;

    const float* Wp = W1 + (size_t)o * EMBED * EMBED;

    v8f acc[6];
    for (int i = 0; i < 6; ++i) acc[i] = (v8f){};

    for (int k0 = 0; k0 < EMBED; k0 += KC) {
        // ---- stage f[:, k0:k0+64] into LDS (all 256 threads) ----
        {
            const int row = t >> 3;           // 0..31
            const int kp  = (t & 7) * 8;      // 0..56
            const float* src = f + row * EMBED + k0 + kp;
            float4 d0 = *(const float4*)(src);
            float4 d1 = *(const float4*)(src + 4);
            float* dst = &fchunk[row * KCP + kp];
            ((float2*)dst)[0] = make_float2(d0.x, d0.y);
            ((float2*)dst)[1] = make_float2(d0.z, d0.w);
            ((float2*)dst)[2] = make_float2(d1.x, d1.y);
            ((float2*)dst)[3] = make_float2(d1.z, d1.w);
        }
        __syncthreads();

        // ---- 16 WMMA K-steps of 4 over this chunk ----
        for (int kk = 0; kk < KC; kk += 4) {
            // A fragment (16x4 f32): VGPR j, half-wave hi -> K = kk + j + 2*hi
            const float* ap = &fchunk[(mtile * 16 + lo) * KCP + kk + 2 * hi];
            v2f a = *(const v2f*)ap;

            // B fragments (4x16 f32): VGPR j, half-wave hi -> K row k0+kk+j+2*hi
            const float* bp = Wp + (size_t)(k0 + kk + 2 * hi) * EMBED + colglb + lo;
            #pragma unroll
            for (int nt = 0; nt < 6; ++nt) {
                v2f b;
                b.x = bp[nt * 16];
                b.y = bp[nt * 16 + EMBED];
                acc[nt] = __builtin_amdgcn_wmma_f32_16x16x4_f32(
                    false, a, false, b, (short)0, acc[nt], false, false);
            }
        }
        __syncthreads();
    }

    // ---- bias + ReLU, write h tile to LDS ----
    #pragma unroll
    for (int nt = 0; nt < 6; ++nt) {
        const int lcol = colloc + nt * 16 + lo;
        const float bias = b1[o * EMBED + colglb + nt * 16 + lo];
        #pragma unroll
        for (int r = 0; r < 8; ++r) {
            // C/D layout: VGPR r -> M = r + 8*hi (within 16-row tile)
            const int m = mtile * 16 + hi * 8 + r;
            float v = acc[nt][r] + bias;
            v = v > 0.0f ? v : 0.0f;
            hbuf[m * HSTR + lcol] = v;
        }
    }
    __syncthreads();

    // ---- layer 2 partial: y_part[b][d] = sum_k h[b,k] * W2[o][nh*384+k][d] ----
    if (t < BATCH * OUTF) {
        const int b = t / OUTF;
        const int d = t % OUTF;
        const float* w2p = W2 + ((size_t)o * EMBED + nh * NHALF) * OUTF + d;
        const float* hp  = &hbuf[b * HSTR];
        float s = 0.0f;
        for (int k = 0; k < NHALF; ++k)
            s += hp[k] * w2p[k * OUTF];
        ws[((size_t)(nh * BATCH + b) * OUTF + d) * HEADS + o] = s;
    }
}

// ---------------------------------------------------------------------------
// Kernel B: out[b][d][o] = ws[0][b][d][o] + ws[1][b][d][o] + b2[o][d]
// ---------------------------------------------------------------------------
__global__ void head_reduce_kernel(const float* __restrict__ ws,
                                   const float* __restrict__ b2,
                                   float* __restrict__ out) {
    const int i = blockIdx.x * 256 + threadIdx.x;
    if (i >= BATCH * OUTF * HEADS) return;
    const int o = i % HEADS;
    const int d = (i / HEADS) % OUTF;
    const int b = i / (HEADS * OUTF);
    const float v0 = ws[((size_t)(b) * OUTF + d) * HEADS + o];
    const float v1 = ws[((size_t)(BATCH + b) * OUTF + d) * HEADS + o];
    out[i] = v0 + v1 + b2[o * OUTF + d];
}

extern "C" void kernel_launch(void* const* d_in, const int* in_sizes, int n_in,
                              void* d_out, int out_size, void* d_ws, size_t ws_size,
                              hipStream_t stream) {
    const float* f  = (const float*)d_in[0];
    const float* W1 = (const float*)d_in[1];
    const float* b1 = (const float*)d_in[2];
    const float* W2 = (const float*)d_in[3];
    const float* b2 = (const float*)d_in[4];
    float* out = (float*)d_out;
    float* ws  = (float*)d_ws;   // needs 2*32*3*300*4 = 230400 bytes

    dim3 gridA(2, HEADS);
    head_mlp_kernel<<<gridA, 256, 0, stream>>>(f, W1, b1, W2, ws);

    const int n = BATCH * OUTF * HEADS;
    head_reduce_kernel<<<(n + 255) / 256, 256, 0, stream>>>(ws, b2, out);
}